// GraphUNet_25950192403208
// MI455X (gfx1250) — compile-verified
//
#include <hip/hip_runtime.h>
#include <hip/hip_bf16.h>
#include <stdint.h>

// Problem constants (match reference)
#define GN0 200000
#define GN1 50000
#define GN2 12500
#define GE0 3200000
#define GE1 800000
#define GE2 200000
#define GC  128

typedef float v2f __attribute__((ext_vector_type(2)));
typedef float v8f __attribute__((ext_vector_type(8)));

static inline unsigned cdiv_u(long long a, long long b) { return (unsigned)((a + b - 1) / b); }

// ---------------- utility kernels ----------------

__global__ void fill_k(float* __restrict__ p, float v, long long n) {
  long long t = (long long)blockIdx.x * blockDim.x + threadIdx.x;
  if (t < n) p[t] = v;
}

__global__ void deg_k(const int* __restrict__ dst, long long e, float* __restrict__ deg) {
  long long t = (long long)blockIdx.x * blockDim.x + threadIdx.x;
  if (t < e) atomicAdd(deg + dst[t], 1.0f);
}

__global__ void rsqrt_k(float* __restrict__ p, long long n) {
  long long t = (long long)blockIdx.x * blockDim.x + threadIdx.x;
  if (t < n) p[t] = rsqrtf(p[t]);   // deg >= 1 always (self loop) -> max(deg,1)==deg
}

// ---------------- fp32 WMMA GEMM: H[n x 128] = X[n x K] @ W[K x 128] ----------------
// One wave computes a 16-row tile across all 128 output columns (8 N-tiles).
// Guard-free: ragged tails (only level 2, n=12500) read/write pad rows that
// stay inside our workspace allocation and are never consumed downstream.
// V_WMMA_F32_16X16X4_F32 operand layouts (ISA 7.12.2):
//   A 16x4 : lane m (0-15): {K=0,K=1}; lane m+16: {K=2,K=3}
//   B 4x16 : lane n (0-15): {K=0,K=1} of column n; lane n+16: {K=2,K=3}
//   D 16x16: VGPR r: lanes 0-15 -> M=r, N=lane; lanes 16-31 -> M=8+r, N=lane-16
template <int K>
__global__ __launch_bounds__(256) void gemm_wmma_k(const float* __restrict__ X,
                                                   const float* __restrict__ W,
                                                   float* __restrict__ H, int ntiles) {
  const int lane = threadIdx.x & 31;
  const int wave = threadIdx.x >> 5;
  const int tile = blockIdx.x * 8 + wave;     // 16-row tile of nodes (wave-uniform)
  if (tile >= ntiles) return;
  const int row0 = tile * 16;
  const int rl   = lane & 15;
  const int hi   = lane >> 4;                 // 0: lanes 0-15, 1: lanes 16-31

  const float* Xrow = X + (long long)(row0 + rl) * K + hi * 2;   // 8B-aligned

  for (int nt = 0; nt < 8; ++nt) {
    const int col = nt * 16 + rl;
    const float* Wc = W + col + hi * 2 * GC;
    v8f acc = {};
#pragma unroll
    for (int k0 = 0; k0 < K; k0 += 4) {
      const v2f a = *(const v2f*)(Xrow + k0);
      v2f b;
      b[0] = Wc[k0 * GC];
      b[1] = Wc[k0 * GC + GC];
      acc = __builtin_amdgcn_wmma_f32_16x16x4_f32(false, a, false, b,
                                                  (short)0, acc, false, false);
    }
    float* Hb = H + (long long)(row0 + hi * 8) * GC + nt * 16 + rl;
#pragma unroll
    for (int r = 0; r < 8; ++r) Hb[r * GC] = acc[r];
  }
}

// ---------------- edge aggregation: agg[d] += h[s] * inv[s]*inv[d] ----------------
// One wave per edge (wave-uniform edge index -> scalar loads of s/d/inv);
// lane handles 4 channels (float4). Feature matrices are L2-resident
// (<=102MB vs 192MB L2), so gathers/atomics mostly hit L2.
__global__ __launch_bounds__(256) void aggregate_k(const int* __restrict__ src,
                                                   const int* __restrict__ dst,
                                                   long long e, const float* __restrict__ H,
                                                   const float* __restrict__ inv,
                                                   float* __restrict__ agg) {
  const long long ed = (long long)blockIdx.x * 8 + (threadIdx.x >> 5);
  const int lane = threadIdx.x & 31;
  if (ed >= e) return;
  const int s = src[ed];
  const int d = dst[ed];
  const float w = inv[s] * inv[d];
  const float4 v = *(const float4*)(H + (long long)s * GC + lane * 4);
  float* o = agg + (long long)d * GC + lane * 4;
  atomicAdd(o + 0, v.x * w);
  atomicAdd(o + 1, v.y * w);
  atomicAdd(o + 2, v.z * w);
  atomicAdd(o + 3, v.w * w);
}

// out = relu(agg + h*inv^2 + bias) ; in-place on agg buffer (float4 per thread)
__global__ void finalize_k(const float* __restrict__ H, const float* __restrict__ inv,
                           const float* __restrict__ bias, float* __restrict__ io,
                           long long nC4) {            // nC4 = n*128/4
  long long t = (long long)blockIdx.x * blockDim.x + threadIdx.x;
  if (t >= nC4) return;
  const long long i = t >> 5;                 // node (32 float4 per row)
  const int c4 = (int)(t & 31);
  const float iv = inv[i];
  const float s2 = iv * iv;
  const float4 a = ((const float4*)io)[t];
  const float4 h = ((const float4*)H)[t];
  const float4 b = ((const float4*)bias)[c4];
  float4 r;
  r.x = fmaxf(a.x + h.x * s2 + b.x, 0.0f);
  r.y = fmaxf(a.y + h.y * s2 + b.y, 0.0f);
  r.z = fmaxf(a.z + h.z * s2 + b.z, 0.0f);
  r.w = fmaxf(a.w + h.w * s2 + b.w, 0.0f);
  ((float4*)io)[t] = r;
}

// cluster max-pool of non-negative (post-ReLU) features; pooled pre-zeroed.
// int atomicMax on float bit patterns is order-preserving for x >= 0, and
// empty clusters stay 0 (matching segment_max + neginf->0).
__global__ void pool_max_k(const float* __restrict__ x, const int* __restrict__ cl,
                           float* __restrict__ p, long long nC) {
  long long t = (long long)blockIdx.x * blockDim.x + threadIdx.x;
  if (t >= nC) return;
  const long long i = t >> 7;
  const int c = (int)(t & 127);
  const int g = cl[i];
  atomicMax((int*)(p + (long long)g * GC + c), __float_as_int(x[t]));
}

// U[i] = concat(up[clusters[i]], skip[i])  -> n x 256  (float4 per thread)
__global__ void gather_concat_k(const float* __restrict__ up, const int* __restrict__ cl,
                                const float* __restrict__ skip, float* __restrict__ U,
                                long long n64) {        // n64 = n*256/4
  long long t = (long long)blockIdx.x * blockDim.x + threadIdx.x;
  if (t >= n64) return;
  const long long i = t >> 6;                 // node (64 float4 per concat row)
  const int q = (int)(t & 63);
  float4 v;
  if (q < 32) v = ((const float4*)(up + (long long)cl[i] * GC))[q];
  else        v = ((const float4*)(skip + i * GC))[q - 32];
  ((float4*)U)[t] = v;
}

// out[i] = dl0[i,:] . w_out + b_out    (wave per node, shuffle reduction)
__global__ void head_k(const float* __restrict__ X, const float* __restrict__ w,
                       const float* __restrict__ b, float* __restrict__ out, int n) {
  long long t = (long long)blockIdx.x * blockDim.x + threadIdx.x;
  const long long node = t >> 5;
  const int lane = (int)(t & 31);
  if (node >= n) return;
  const float4 xv = *(const float4*)(X + node * GC + lane * 4);
  const float4 wv = *(const float4*)(w + lane * 4);
  float s = xv.x * wv.x + xv.y * wv.y + xv.z * wv.z + xv.w * wv.w;
  for (int off = 16; off; off >>= 1) s += __shfl_down(s, off, 32);
  if (lane == 0) out[node] = s + b[0];
}

// ---------------- host-side layer driver ----------------

static void launch_gemm(const float* X, int K, const float* W, float* H, int n,
                        hipStream_t stream) {
  const int tiles = (int)cdiv_u(n, 16);
  const unsigned grid = cdiv_u(tiles, 8);
  if (K == 4)
    gemm_wmma_k<4><<<grid, 256, 0, stream>>>(X, W, H, tiles);
  else if (K == 128)
    gemm_wmma_k<128><<<grid, 256, 0, stream>>>(X, W, H, tiles);
  else
    gemm_wmma_k<256><<<grid, 256, 0, stream>>>(X, W, H, tiles);
}

// relu(gcn(X, W, b)) -> out ; X, H, out must be distinct buffers
static void gcn_layer(const float* X, int K, const float* W, const float* bias,
                      const int* src, const int* dst, long long E,
                      const float* inv, int n, float* H, float* out,
                      hipStream_t stream) {
  launch_gemm(X, K, W, H, n, stream);
  const long long nC = (long long)n * GC;
  fill_k<<<cdiv_u(nC, 256), 256, 0, stream>>>(out, 0.0f, nC);
  aggregate_k<<<cdiv_u(E, 8), 256, 0, stream>>>(src, dst, E, H, inv, out);
  finalize_k<<<cdiv_u(nC / 4, 256), 256, 0, stream>>>(H, inv, bias, out, nC / 4);
}

extern "C" void kernel_launch(void* const* d_in, const int* in_sizes, int n_in,
                              void* d_out, int out_size, void* d_ws, size_t ws_size,
                              hipStream_t stream) {
  (void)in_sizes; (void)n_in; (void)out_size; (void)ws_size;

  const float* x    = (const float*)d_in[0];
  const int* ei0    = (const int*)d_in[1];
  const int* ei1    = (const int*)d_in[2];
  const int* ei2    = (const int*)d_in[3];
  const int* cl0    = (const int*)d_in[4];
  const int* cl1    = (const int*)d_in[5];
  const float *w_e0a = (const float*)d_in[6],  *b_e0a = (const float*)d_in[7];
  const float *w_e0b = (const float*)d_in[8],  *b_e0b = (const float*)d_in[9];
  const float *w_e1a = (const float*)d_in[10], *b_e1a = (const float*)d_in[11];
  const float *w_e1b = (const float*)d_in[12], *b_e1b = (const float*)d_in[13];
  const float *w_ba  = (const float*)d_in[14], *b_ba  = (const float*)d_in[15];
  const float *w_bb  = (const float*)d_in[16], *b_bb  = (const float*)d_in[17];
  const float *w_d1a = (const float*)d_in[18], *b_d1a = (const float*)d_in[19];
  const float *w_d1b = (const float*)d_in[20], *b_d1b = (const float*)d_in[21];
  const float *w_d0a = (const float*)d_in[22], *b_d0a = (const float*)d_in[23];
  const float *w_d0b = (const float*)d_in[24], *b_d0b = (const float*)d_in[25];
  const float *w_out = (const float*)d_in[26], *b_out = (const float*)d_in[27];

  const int* src0 = ei0;        const int* dst0 = ei0 + GE0;
  const int* src1 = ei1;        const int* dst1 = ei1 + GE1;
  const int* src2 = ei2;        const int* dst2 = ei2 + GE2;

  // ---- workspace carve (floats) ----
  float* ws = (float*)d_ws;
  float* inv0 = ws;                       ws += GN0;
  float* inv1 = ws;                       ws += GN1;
  float* inv2 = ws;                       ws += GN2;
  float* R0   = ws;                       ws += (long long)GN0 * GC;
  float* R1   = ws;                       ws += (long long)GN0 * GC;
  float* R2   = ws;                       ws += (long long)GN0 * GC;
  float* e0   = ws;                       ws += (long long)GN0 * GC;
  float* e1   = ws;                       ws += (long long)GN1 * GC;
  float* U    = ws;                       ws += (long long)GN0 * 2 * GC;

  // ---- degrees & symmetric normalization (once per level) ----
  fill_k<<<cdiv_u(GN0, 256), 256, 0, stream>>>(inv0, 1.0f, GN0);   // self loop
  fill_k<<<cdiv_u(GN1, 256), 256, 0, stream>>>(inv1, 1.0f, GN1);
  fill_k<<<cdiv_u(GN2, 256), 256, 0, stream>>>(inv2, 1.0f, GN2);
  deg_k<<<cdiv_u(GE0, 256), 256, 0, stream>>>(dst0, GE0, inv0);
  deg_k<<<cdiv_u(GE1, 256), 256, 0, stream>>>(dst1, GE1, inv1);
  deg_k<<<cdiv_u(GE2, 256), 256, 0, stream>>>(dst2, GE2, inv2);
  rsqrt_k<<<cdiv_u(GN0, 256), 256, 0, stream>>>(inv0, GN0);
  rsqrt_k<<<cdiv_u(GN1, 256), 256, 0, stream>>>(inv1, GN1);
  rsqrt_k<<<cdiv_u(GN2, 256), 256, 0, stream>>>(inv2, GN2);

  // ---- encoder level 0 ----
  gcn_layer(x,  4,  w_e0a, b_e0a, src0, dst0, GE0, inv0, GN0, R0, R1, stream);
  gcn_layer(R1, GC, w_e0b, b_e0b, src0, dst0, GE0, inv0, GN0, R0, e0, stream);

  // ---- pool 0 -> level 1 ----
  fill_k<<<cdiv_u((long long)GN1 * GC, 256), 256, 0, stream>>>(R0, 0.0f, (long long)GN1 * GC);
  pool_max_k<<<cdiv_u((long long)GN0 * GC, 256), 256, 0, stream>>>(e0, cl0, R0, (long long)GN0 * GC);

  // ---- encoder level 1 ----
  gcn_layer(R0, GC, w_e1a, b_e1a, src1, dst1, GE1, inv1, GN1, R1, R2, stream);
  gcn_layer(R2, GC, w_e1b, b_e1b, src1, dst1, GE1, inv1, GN1, R1, e1, stream);

  // ---- pool 1 -> level 2 ----
  fill_k<<<cdiv_u((long long)GN2 * GC, 256), 256, 0, stream>>>(R0, 0.0f, (long long)GN2 * GC);
  pool_max_k<<<cdiv_u((long long)GN1 * GC, 256), 256, 0, stream>>>(e1, cl1, R0, (long long)GN1 * GC);

  // ---- bottom level 2 ----
  gcn_layer(R0, GC, w_ba, b_ba, src2, dst2, GE2, inv2, GN2, R1, R2, stream);
  gcn_layer(R2, GC, w_bb, b_bb, src2, dst2, GE2, inv2, GN2, R1, R0, stream);  // bt = R0

  // ---- unpool -> level 1, concat skip ----
  gather_concat_k<<<cdiv_u((long long)GN1 * 64, 256), 256, 0, stream>>>(
      R0, cl1, e1, U, (long long)GN1 * 64);

  // ---- decoder level 1 ----
  gcn_layer(U,  2 * GC, w_d1a, b_d1a, src1, dst1, GE1, inv1, GN1, R1, R2, stream);
  gcn_layer(R2, GC,     w_d1b, b_d1b, src1, dst1, GE1, inv1, GN1, R1, R0, stream); // dl1 = R0

  // ---- unpool -> level 0, concat skip ----
  gather_concat_k<<<cdiv_u((long long)GN0 * 64, 256), 256, 0, stream>>>(
      R0, cl0, e0, U, (long long)GN0 * 64);

  // ---- decoder level 0 ----
  gcn_layer(U,  2 * GC, w_d0a, b_d0a, src0, dst0, GE0, inv0, GN0, R1, R2, stream);
  gcn_layer(R2, GC,     w_d0b, b_d0b, src0, dst0, GE0, inv0, GN0, R1, R0, stream); // dl0 = R0

  // ---- output head ----
  head_k<<<cdiv_u((long long)GN0 * 32, 256), 256, 0, stream>>>(
      R0, w_out, b_out, (float*)d_out, GN0);
}